// GIN_quan_60266981098197
// MI455X (gfx1250) — compile-verified
//
#include <hip/hip_runtime.h>

#define D 128
#define NGRAPH 128

typedef __attribute__((ext_vector_type(2))) float v2f;
typedef __attribute__((ext_vector_type(8))) float v8f;

// ---------------------------------------------------------------------------
// Per-tensor symmetric int4 fake-quant: s = max|W|/7 ; q = clip(rint(W/s),-8,7)*s
// One block (256 threads) per 128x128 weight matrix.
// ---------------------------------------------------------------------------
__global__ void quant4_kernel(const float* __restrict__ W, float* __restrict__ Q) {
  __shared__ float red[256];
  float m = 0.0f;
  for (int i = threadIdx.x; i < D * D; i += 256) m = fmaxf(m, fabsf(W[i]));
  red[threadIdx.x] = m;
  __syncthreads();
  for (int s = 128; s > 0; s >>= 1) {
    if (threadIdx.x < s) red[threadIdx.x] = fmaxf(red[threadIdx.x], red[threadIdx.x + s]);
    __syncthreads();
  }
  float scale = fmaxf(red[0] / 7.0f, 1e-30f);
  float inv = 1.0f / scale;
  for (int i = threadIdx.x; i < D * D; i += 256) {
    float q = rintf(W[i] * inv);          // round half to even, matches jnp.round
    q = fminf(fmaxf(q, -8.0f), 7.0f);
    Q[i] = q * scale;
  }
}

__global__ void zero_kernel(float* __restrict__ p, int n) {
  int i = blockIdx.x * blockDim.x + threadIdx.x;
  if (i < n) p[i] = 0.0f;
}

__global__ void copy4_kernel(const float4* __restrict__ in, float4* __restrict__ out, long n4) {
  long i = blockIdx.x * (long)blockDim.x + threadIdx.x;
  if (i < n4) out[i] = in[i];
}

// ---------------------------------------------------------------------------
// Edge scatter-add: one wave per edge, lane handles 4 consecutive features.
// T[dst] += H[src]  (float atomics resolve in the 192MB L2; h is 25.6MB)
// ---------------------------------------------------------------------------
__global__ void scatter_add_kernel(const float* __restrict__ H,
                                   const int* __restrict__ srcv,
                                   const int* __restrict__ dstv,
                                   float* __restrict__ T, int E) {
  long t = blockIdx.x * (long)blockDim.x + threadIdx.x;
  int e = (int)(t >> 5);
  int lane = (int)(t & 31);
  if (e >= E) return;
  int s = srcv[e];
  int d = dstv[e];
  const float4 v = *(const float4*)(H + (long)s * D + lane * 4);
  float* p = T + (long)d * D + lane * 4;
  unsafeAtomicAdd(p + 0, v.x);
  unsafeAtomicAdd(p + 1, v.y);
  unsafeAtomicAdd(p + 2, v.z);
  unsafeAtomicAdd(p + 3, v.w);
}

// ---------------------------------------------------------------------------
// Fused 2-layer MLP on a 16-row strip, fp32 WMMA 16x16x4.
//   h_mid = relu(A @ Wa^T + ba)   -> staged in LDS (stride 132, conflict-free)
//   out   = h_mid @ Wb^T + bb
// Block = 256 threads = 8 waves; wave w owns column tile w.
// mode 0: O[row] = relu(out)      (inter-layer)
// mode 1: sums[batch[row]] += out (fused global mean-pool accumulate, last layer)
// ---------------------------------------------------------------------------
__global__ void mlp_fused_kernel(const float* __restrict__ A,
                                 const float* __restrict__ Wa,
                                 const float* __restrict__ ba,
                                 const float* __restrict__ Wb,
                                 const float* __restrict__ bb,
                                 float* __restrict__ O,
                                 const int* __restrict__ batch,
                                 float* __restrict__ sums,
                                 int N, int mode) {
  __shared__ float smem[16][132];            // 16 rows of h_mid, padded stride

  const int wave = threadIdx.x >> 5;
  const int lane = threadIdx.x & 31;
  const int half = lane >> 4;                // 0: K pair {0,1}; 1: K pair {2,3}
  const int l16  = lane & 15;
  const int rowBase = blockIdx.x * 16;
  const int col = wave * 16 + l16;

  int arow = rowBase + l16;
  if (arow > N - 1) arow = N - 1;            // branch-free clamp, EXEC stays full
  const float* ap  = A + (long)arow * D + 2 * half;
  const float* bpa = Wa + (long)col * D + 2 * half;
  const float* bpb = Wb + (long)col * D + 2 * half;

  // ---- phase 1: h_mid tile = relu(A @ Wa^T + ba) -> LDS ----
  v8f c = {};
#pragma unroll
  for (int k0 = 0; k0 < D; k0 += 4) {
    v2f a = *(const v2f*)(ap + k0);
    v2f b = *(const v2f*)(bpa + k0);
    c = __builtin_amdgcn_wmma_f32_16x16x4_f32(false, a, false, b, (short)0, c,
                                              false, false);
  }
  {
    const float bv = ba[col];
#pragma unroll
    for (int i = 0; i < 8; ++i) {
      smem[i + 8 * half][col] = fmaxf(c[i] + bv, 0.0f);   // inner-MLP relu
    }
  }
  __syncthreads();

  // ---- phase 2: out tile = h_mid @ Wb^T + bb, A fragments from LDS ----
  v8f c2 = {};
#pragma unroll
  for (int k0 = 0; k0 < D; k0 += 4) {
    v2f a = *(const v2f*)(&smem[l16][k0 + 2 * half]);     // ds_load_b64, no conflicts
    v2f b = *(const v2f*)(bpb + k0);
    c2 = __builtin_amdgcn_wmma_f32_16x16x4_f32(false, a, false, b, (short)0, c2,
                                               false, false);
  }

  const float bv = bb[col];
  if (mode == 0) {                           // inter-layer: write relu(out)
#pragma unroll
    for (int i = 0; i < 8; ++i) {
      int r = rowBase + i + 8 * half;        // C/D layout: VGPR i -> M = i + 8*half
      if (r < N) O[(long)r * D + col] = fmaxf(c2[i] + bv, 0.0f);
    }
  } else {                                   // last layer: fused pool accumulate
#pragma unroll
    for (int i = 0; i < 8; ++i) {
      int r = rowBase + i + 8 * half;
      if (r < N) {
        int g = batch[r];
        unsafeAtomicAdd(sums + (long)g * D + col, c2[i] + bv);
      }
    }
  }
}

// ---------------------------------------------------------------------------
// Per-graph node counts for mean pooling.
// ---------------------------------------------------------------------------
__global__ void count_kernel(const int* __restrict__ batch, float* __restrict__ counts,
                             int N) {
  int i = blockIdx.x * blockDim.x + threadIdx.x;
  if (i < N) unsafeAtomicAdd(counts + batch[i], 1.0f);
}

__global__ void pool_final_kernel(const float* __restrict__ sums,
                                  const float* __restrict__ counts,
                                  float* __restrict__ out) {
  int t = blockIdx.x * blockDim.x + threadIdx.x;   // NGRAPH*D = 16384 threads
  int g = t >> 7;
  out[t] = sums[t] / fmaxf(counts[g], 1.0f);
}

// ---------------------------------------------------------------------------
extern "C" void kernel_launch(void* const* d_in, const int* in_sizes, int n_in,
                              void* d_out, int out_size, void* d_ws, size_t ws_size,
                              hipStream_t stream) {
  const float* x     = (const float*)d_in[0];
  const int*   ei    = (const int*)d_in[1];
  const int*   batch = (const int*)d_in[2];
  const int N = in_sizes[0] / D;
  const int E = in_sizes[1] / 2;
  const int* srcv = ei;
  const int* dstv = ei + E;

  float* ws     = (float*)d_ws;
  float* qw     = ws;                        // 6 * 16384 floats
  float* sums   = ws + 6 * D * D;            // 16384 floats
  float* counts = sums + NGRAPH * D;         // 128 floats
  float* bufT   = ws + 131072;               // [N, D]  (h + aggregation)
  float* bufH   = bufT + (size_t)N * D;      // [N, D]  (layer output)

  // 1) quantize the six weight matrices (w{l}a at 3+4l, w{l}b at 5+4l)
  for (int l = 0; l < 3; ++l) {
    quant4_kernel<<<1, 256, 0, stream>>>((const float*)d_in[3 + 4 * l],
                                         qw + (size_t)(2 * l) * D * D);
    quant4_kernel<<<1, 256, 0, stream>>>((const float*)d_in[5 + 4 * l],
                                         qw + (size_t)(2 * l + 1) * D * D);
  }

  // 2) zero pooling accumulators (every call; harness never re-poisons ws)
  zero_kernel<<<(NGRAPH * D + NGRAPH + 255) / 256, 256, 0, stream>>>(
      sums, NGRAPH * D + NGRAPH);
  // node counts per graph (independent of the layer pipeline)
  count_kernel<<<(N + 255) / 256, 256, 0, stream>>>(batch, counts, N);

  const long n4 = (long)N * D / 4;
  const int copyBlocks = (int)((n4 + 255) / 256);
  const int scatBlocks = (int)(((long)E * 32 + 255) / 256);
  const int gemmBlocks = (N + 15) / 16;

  const float* hin = x;
  for (int l = 0; l < 3; ++l) {
    // tmp = h  (the (1+eps)*x term, eps=0)
    copy4_kernel<<<copyBlocks, 256, 0, stream>>>((const float4*)hin, (float4*)bufT, n4);
    // tmp[dst] += h[src]
    scatter_add_kernel<<<scatBlocks, 256, 0, stream>>>(hin, srcv, dstv, bufT, E);
    // fused 2-layer MLP; last layer accumulates straight into pool sums
    mlp_fused_kernel<<<gemmBlocks, 256, 0, stream>>>(
        bufT,
        qw + (size_t)(2 * l) * D * D,     (const float*)d_in[4 + 4 * l],
        qw + (size_t)(2 * l + 1) * D * D, (const float*)d_in[6 + 4 * l],
        bufH, batch, sums, N, (l < 2) ? 0 : 1);
    hin = bufH;
  }

  // 3) finalize mean pooling into d_out (writes every output element)
  pool_final_kernel<<<(NGRAPH * D) / 256, 256, 0, stream>>>(sums, counts,
                                                            (float*)d_out);
}